// GCNRecommendationModel_26852135535045
// MI455X (gfx1250) — compile-verified
//
#include <hip/hip_runtime.h>
#include <hip/hip_bf16.h>

typedef float v2f __attribute__((ext_vector_type(2)));
typedef float v8f __attribute__((ext_vector_type(8)));

#define F_IN 22    // layer-1 input feature count
#define HID  64    // hidden width

// ---------------- degree / normalization ----------------

__global__ void k_init_deg(float* __restrict__ deg, int n) {
    int i = blockIdx.x * blockDim.x + threadIdx.x;
    if (i < n) deg[i] = 1.0f;                 // self-loop contributes 1
}

__global__ void k_count_deg(const int* __restrict__ dst, float* __restrict__ deg, int e) {
    int i = blockIdx.x * blockDim.x + threadIdx.x;
    if (i < e) atomicAdd(&deg[dst[i]], 1.0f);
}

__global__ void k_rsqrt(float* __restrict__ deg, int n) {
    int i = blockIdx.x * blockDim.x + threadIdx.x;
    if (i < n) deg[i] = rsqrtf(deg[i]);       // deg >= 1 always (self-loop)
}

// ---------------- fused bias + ReLU (in place) ----------------

__global__ void k_bias_relu(float* __restrict__ h, const float* __restrict__ b, long long n64) {
    long long gid = (long long)blockIdx.x * blockDim.x + threadIdx.x;
    if (gid < n64) h[gid] = fmaxf(h[gid] + b[(int)(gid & (HID - 1))], 0.0f);
}

// ---------------- dense transform via V_WMMA_F32_16X16X4_F32 ----------------
// 256-thread block = 8 waves; each wave computes a 16-row x 64-col output tile.
// W is staged in LDS in K-PAIR-INTERLEAVED layout:
//     Ws[((k/2)*64 + c)*2 + (k&1)] = W[k][c]   (zero-padded to KPAD rows)
// so a B fragment {W[ka][c], W[ka+1][c]} (ka even) is ONE aligned ds_load_b64
// landing directly in an even VGPR pair for the WMMA src1 operand.
// A fragment (16x4 f32): lanes 0-15 hold K=k0..k0+1, lanes 16-31 hold K=k0+2..k0+3.
// C/D (16x16 f32): 8 VGPRs, M = r + 8*(lane>=16), N = lane&15.
template <int K, int KPAD>
__global__ void k_transform_wmma(const float* __restrict__ in, int ldin,
                                 const float* __restrict__ W,   // K x 64, row-major
                                 float* __restrict__ out,       // n x 64
                                 int n)
{
    __shared__ float Ws[KPAD * HID];
    for (int idx = threadIdx.x; idx < (KPAD / 2) * HID; idx += blockDim.x) {
        const int k2 = idx / HID;            // K-pair index
        const int c  = idx - k2 * HID;       // column
        const int k  = 2 * k2;
        float w0 = (k     < K) ? W[(size_t)k       * HID + c] : 0.0f;
        float w1 = (k + 1 < K) ? W[(size_t)(k + 1) * HID + c] : 0.0f;
        Ws[idx * 2 + 0] = w0;
        Ws[idx * 2 + 1] = w1;
    }
    __syncthreads();

    const int wave = threadIdx.x >> 5;               // 0..7
    const int lane = threadIdx.x & 31;
    const int row0 = (blockIdx.x * 8 + wave) * 16;
    const int half = lane >> 4;                      // 0 or 1
    const int mr   = lane & 15;
    const int arow = min(row0 + mr, n - 1);          // clamp: keep EXEC full
    const float* __restrict__ inrow = in + (size_t)arow * ldin;
    const v2f* __restrict__ Wp = (const v2f*)Ws;     // pair view

    v8f acc[4] = {};                                 // 4 N-tiles -> 64 cols/row

    #pragma unroll
    for (int k0 = 0; k0 < KPAD; k0 += 4) {
        const int ka = k0 + half * 2;                // always even
        v2f a;
        if (k0 + 4 <= K) {                           // compile-time: clean b64 load
            float2 t = *(const float2*)(inrow + ka);
            a.x = t.x; a.y = t.y;
        } else {                                     // only the padded tail chunk
            a.x = (ka     < K) ? inrow[ka]     : 0.0f;
            a.y = (ka + 1 < K) ? inrow[ka + 1] : 0.0f;
        }
        const v2f* wk = Wp + (ka >> 1) * HID + mr;   // one ds_load_b64 per fragment
        #pragma unroll
        for (int nt = 0; nt < 4; ++nt) {
            v2f b = wk[nt * 16];
            acc[nt] = __builtin_amdgcn_wmma_f32_16x16x4_f32(
                false, a, false, b, (short)0, acc[nt], false, false);
        }
    }

    #pragma unroll
    for (int nt = 0; nt < 4; ++nt) {
        #pragma unroll
        for (int r = 0; r < 8; ++r) {
            int m = row0 + half * 8 + r;
            if (m < n) out[(size_t)m * HID + nt * 16 + mr] = acc[nt][r];
        }
    }
}

// ---------------- aggregation ----------------

// agg[i] = dinv[i]^2 * t[i]   (self-loop message; also initializes agg)
__global__ void k_selfloop_init(const float* __restrict__ t, const float* __restrict__ dinv,
                                float* __restrict__ agg, int n) {
    long long gid = (long long)blockIdx.x * blockDim.x + threadIdx.x;
    int i = (int)(gid >> 6), f = (int)(gid & 63);
    if (i < n) {
        float d = dinv[i];
        agg[(size_t)i * HID + f] = d * d * t[(size_t)i * HID + f];
    }
}

// agg[dst] += dinv[src]*dinv[dst] * t[src]
// 16 threads per edge, float4 gather (global_load_b128), 4 atomics per thread.
__global__ void k_scatter(const int* __restrict__ src, const int* __restrict__ dst,
                          const float* __restrict__ dinv, const float* __restrict__ t,
                          float* __restrict__ agg, int e) {
    long long gid = (long long)blockIdx.x * blockDim.x + threadIdx.x;
    int ei = (int)(gid >> 4), f4 = (int)(gid & 15) * 4;
    if (ei < e) {
        int s = src[ei], d = dst[ei];
        float w = dinv[s] * dinv[d];
        float4 v = *(const float4*)(t + (size_t)s * HID + f4);
        float* o = agg + (size_t)d * HID + f4;
        atomicAdd(o + 0, w * v.x);
        atomicAdd(o + 1, w * v.y);
        atomicAdd(o + 2, w * v.z);
        atomicAdd(o + 3, w * v.w);
    }
}

// ---------------- final readout ----------------

__global__ void k_final(const float* __restrict__ agg, const float* __restrict__ b2,
                        const float* __restrict__ Wf, const float* __restrict__ bf,
                        float* __restrict__ out, int n) {
    int i = blockIdx.x * blockDim.x + threadIdx.x;
    if (i >= n) return;
    const float4* row = (const float4*)(agg + (size_t)i * HID);
    const float4* bb4 = (const float4*)b2;
    const float4* wf4 = (const float4*)Wf;
    float acc = 0.0f;
    #pragma unroll
    for (int q = 0; q < HID / 4; ++q) {
        float4 v = row[q], b = bb4[q], w = wf4[q];
        acc += fmaxf(v.x + b.x, 0.0f) * w.x;
        acc += fmaxf(v.y + b.y, 0.0f) * w.y;
        acc += fmaxf(v.z + b.z, 0.0f) * w.z;
        acc += fmaxf(v.w + b.w, 0.0f) * w.w;
    }
    out[i] = acc + bf[0];
}

// ---------------- launch ----------------

extern "C" void kernel_launch(void* const* d_in, const int* in_sizes, int n_in,
                              void* d_out, int out_size, void* d_ws, size_t ws_size,
                              hipStream_t stream) {
    const float* x  = (const float*)d_in[0];
    const int*   ei = (const int*)  d_in[1];
    const float* W1 = (const float*)d_in[2];
    const float* b1 = (const float*)d_in[3];
    const float* W2 = (const float*)d_in[4];
    const float* b2 = (const float*)d_in[5];
    const float* Wf = (const float*)d_in[6];
    const float* bf = (const float*)d_in[7];
    float* out = (float*)d_out;

    const int N = in_sizes[0] / F_IN;        // 100000
    const int E = in_sizes[1] / 2;           // 3200000
    const int* src = ei;
    const int* dst = ei + E;

    // workspace: dinv[N] | bufT[N*64] | bufA[N*64]   (~51.6 MB)
    float* dinv = (float*)d_ws;
    float* bufT = dinv + ((N + 63) & ~63);
    float* bufA = bufT + (size_t)N * HID;

    const int TB = 256;
    const int gN    = (N + TB - 1) / TB;
    const int gE    = (E + TB - 1) / TB;
    const int gN64  = (int)(((long long)N * HID + TB - 1) / TB);
    const int gE16  = (int)(((long long)E * 16  + TB - 1) / TB);
    const int gT    = (N + 127) / 128;       // 8 wave-tiles of 16 rows per block

    // normalization: deg -> dinv
    k_init_deg <<<gN, TB, 0, stream>>>(dinv, N);
    k_count_deg<<<gE, TB, 0, stream>>>(dst, dinv, E);
    k_rsqrt    <<<gN, TB, 0, stream>>>(dinv, N);

    // layer 1: t1 = x @ W1 ; agg1 = D^-1/2 (A+I) D^-1/2 t1
    k_transform_wmma<F_IN, 24><<<gT, TB, 0, stream>>>(x, F_IN, W1, bufT, N);
    k_selfloop_init<<<gN64, TB, 0, stream>>>(bufT, dinv, bufA, N);
    k_scatter      <<<gE16, TB, 0, stream>>>(src, dst, dinv, bufT, bufA, E);

    // layer 2: h = relu(agg1 + b1) ; t2 = h @ W2 ; agg2 = D^-1/2 (A+I) D^-1/2 t2
    k_bias_relu<<<gN64, TB, 0, stream>>>(bufA, b1, (long long)N * HID);
    k_transform_wmma<HID, HID><<<gT, TB, 0, stream>>>(bufA, HID, W2, bufT, N);
    k_selfloop_init<<<gN64, TB, 0, stream>>>(bufT, dinv, bufA, N);
    k_scatter      <<<gE16, TB, 0, stream>>>(src, dst, dinv, bufT, bufA, E);

    // readout: out = relu(agg2 + b2) @ Wf + bf
    k_final<<<gN, TB, 0, stream>>>(bufA, b2, Wf, bf, out, N);
}